// FilterLayer_17652315587294
// MI455X (gfx1250) — compile-verified
//
#include <hip/hip_runtime.h>
#include <hip/hip_bf16.h>

typedef __attribute__((ext_vector_type(2))) float v2f;
typedef __attribute__((ext_vector_type(8))) float v8f;

#define T_LEN   8192
#define N_CH    4096
#define SEG     1024
#define WARM    512
#define NSEG    (T_LEN / SEG)        // 8
#define WBLK    (WARM / 16)          // 32 warm-up blocks
#define NBLK    ((WARM + SEG) / 16)  // 96 blocks per wave
#define XPITCH  68                   // 64 + 4 pad (bank-conflict-free ds loads)
#define WPB     8                    // waves per workgroup
#define CGRPS   (N_CH / 16)          // 256 channel groups

// ---------------------------------------------------------------------------
// Setup: build the block state-space coefficient tables from A, B.
//   ws[0..383]    : Ycoef 16x24 row-major  = [ G(16x8) | H(16x16 lower-Toeplitz) ]
//   ws[384..767]  : Zcoef 16x24 row-major  = [ F^16(8x8) | C(8x16) ; zeros(8x24) ]
// DF2T state space: z' = F z + g x ; y = z[0] + b0 x
//   F[k][j] = (j==k+1) - a[k+1]*(j==0) ;  g[k] = b[k+1] - a[k+1]*b0
// ---------------------------------------------------------------------------
__global__ void iir_setup_kernel(const float* __restrict__ Ain,
                                 const float* __restrict__ Bin,
                                 float* __restrict__ ws) {
  if (threadIdx.x != 0 || blockIdx.x != 0) return;
  float a[9], b[9];
  float inv = 1.0f / Ain[0];
  for (int i = 0; i < 9; ++i) { a[i] = Ain[i] * inv; b[i] = Bin[i] * inv; }
  const float d = b[0];
  float g[8];
  for (int k = 0; k < 8; ++k) g[k] = b[k + 1] - a[k + 1] * d;
  float F[8][8];
  for (int k = 0; k < 8; ++k)
    for (int j = 0; j < 8; ++j)
      F[k][j] = ((j == k + 1) ? 1.0f : 0.0f) - a[k + 1] * ((j == 0) ? 1.0f : 0.0f);
  // Fp[t] = F^t, t = 0..16
  float Fp[17][8][8];
  for (int k = 0; k < 8; ++k)
    for (int j = 0; j < 8; ++j) Fp[0][k][j] = (k == j) ? 1.0f : 0.0f;
  for (int t = 1; t <= 16; ++t)
    for (int k = 0; k < 8; ++k)
      for (int j = 0; j < 8; ++j) {
        float s = 0.0f;
        for (int i = 0; i < 8; ++i) s += F[k][i] * Fp[t - 1][i][j];
        Fp[t][k][j] = s;
      }
  // impulse response h[m]: h[0]=d, h[m]= e0^T F^{m-1} g
  float h[16];
  h[0] = d;
  for (int m = 1; m < 16; ++m) {
    float s = 0.0f;
    for (int j = 0; j < 8; ++j) s += Fp[m - 1][0][j] * g[j];
    h[m] = s;
  }
  float* Yc = ws;
  float* Zc = ws + 16 * 24;
  for (int t = 0; t < 16; ++t) {
    for (int j = 0; j < 8; ++j)  Yc[t * 24 + j] = Fp[t][0][j];        // G
    for (int j = 0; j < 16; ++j) Yc[t * 24 + 8 + j] = (j <= t) ? h[t - j] : 0.0f;
  }
  for (int k = 0; k < 16; ++k)
    for (int c = 0; c < 24; ++c) Zc[k * 24 + c] = 0.0f;
  for (int k = 0; k < 8; ++k) {
    for (int j = 0; j < 8; ++j) Zc[k * 24 + j] = Fp[16][k][j];        // F^16
    for (int j = 0; j < 16; ++j) {                                    // C[:,j] = F^{15-j} g
      float s = 0.0f;
      for (int i = 0; i < 8; ++i) s += Fp[15 - j][k][i] * g[i];
      Zc[k * 24 + 8 + j] = s;
    }
  }
}

// ---------------------------------------------------------------------------
// Main: each wave filters 16 channels over one 1024-sample segment
// (+512 warm-up; segment 0's warm-up reads zeros = exact rest state).
// Per 16-sample block: B=[Z;X] (24x16ch), Z' and Y via 6 WMMA f32 16x16x4 each.
// ---------------------------------------------------------------------------
__global__ __launch_bounds__(256) void iir_wmma_kernel(
    const float* __restrict__ x, const float* __restrict__ ws,
    float* __restrict__ out) {
  __shared__ float smem[WPB * (16 * XPITCH + 16 * 8)];

  const int lane = threadIdx.x & 31;
  const int wid  = threadIdx.x >> 5;
  const int gw   = blockIdx.x * WPB + wid;   // 0..2047
  const int cg   = gw & (CGRPS - 1);
  const int seg  = gw >> 8;
  const int chBase = cg * 16;
  const int t0     = seg * SEG;

  float* Xlds = smem + wid * (16 * XPITCH + 16 * 8);
  float* Zlds = Xlds + 16 * XPITCH;

  const int nIdx = lane & 15;   // channel (B/D column) / coef row (A)
  const int half = lane >> 4;   // K sub-pair select per WMMA layout

  // A operands: [G|H] and [F^16|C], K=24 -> 6 chunks of K=4, v2f each.
  // A 16x4 layout: VGPR0 <-> K{0 lanes0-15, 2 lanes16-31}, VGPR1 <-> K{1,3}.
  const float* Yc = ws;
  const float* Zc = ws + 384;
  v2f aY[6], aZ[6];
#pragma unroll
  for (int c = 0; c < 6; ++c) {
    const int col = 4 * c + 2 * half;
    aY[c] = *(const v2f*)(Yc + nIdx * 24 + col);
    aZ[c] = *(const v2f*)(Zc + nIdx * 24 + col);
  }

  // zero initial state (warm-up converges it; seg 0 exact)
  if (lane < 16) {
#pragma unroll
    for (int k = 0; k < 8; ++k) Zlds[nIdx * 8 + k] = 0.0f;
  }

  for (int blk = 0; blk < NBLK; ++blk) {
    const int tb = blk & 3;
    if (tb == 0) {
      // stage 16ch x 64t tile into LDS (float4 per lane x 8)
      const int tbase = t0 - WARM + blk * 16;
#pragma unroll
      for (int i = 0; i < 8; ++i) {
        const int idx = i * 32 + lane;
        const int ch  = idx >> 4;
        const int tq  = (idx & 15) << 2;
        float4 v = make_float4(0.f, 0.f, 0.f, 0.f);
        if (tbase >= 0)
          v = *(const float4*)(x + (size_t)(chBase + ch) * T_LEN + tbase + tq);
        *(float4*)(Xlds + ch * XPITCH + tq) = v;
      }
      // hint-prefetch the next tile (global_prefetch_b8)
      const int tnext = tbase + 64;
      if (blk + 4 < NBLK && tnext >= 0)
        __builtin_prefetch(x + (size_t)(chBase + (lane >> 1)) * T_LEN + tnext +
                               (lane & 1) * 32, 0, 1);
    }

    // B = [Z ; X] (24 x 16ch): chunk c -> v2f holding K rows {4c+2h, 4c+2h+1}
    v2f Bop[6];
    Bop[0] = *(const v2f*)(Zlds + nIdx * 8 + 2 * half);
    Bop[1] = *(const v2f*)(Zlds + nIdx * 8 + 4 + 2 * half);
    const int tbb = tb * 16 + 2 * half;
#pragma unroll
    for (int c = 0; c < 4; ++c)
      Bop[2 + c] = *(const v2f*)(Xlds + nIdx * XPITCH + tbb + 4 * c);

    // Z' = [F^16 | C] * [Z;X]  (rows 8..15 of A are zero -> D rows 8..15 = 0)
    v8f accZ = {};
#pragma unroll
    for (int c = 0; c < 6; ++c)
      accZ = __builtin_amdgcn_wmma_f32_16x16x4_f32(
          false, aZ[c], false, Bop[c], (short)0, accZ, false, false);

    if (blk >= WBLK) {
      // Y = [G | H] * [Z;X]
      v8f accY = {};
#pragma unroll
      for (int c = 0; c < 6; ++c)
        accY = __builtin_amdgcn_wmma_f32_16x16x4_f32(
            false, aY[c], false, Bop[c], (short)0, accY, false, false);
      // clamp to [-1,1] and store: lane holds y[t+v] (v=0..7), t offset +8 for upper half
      const int tglob = t0 + (blk - WBLK) * 16 + half * 8;
      float* o = out + (size_t)(chBase + nIdx) * T_LEN + tglob;
      float4 lo, hi;
      lo.x = fminf(fmaxf(accY[0], -1.0f), 1.0f);
      lo.y = fminf(fmaxf(accY[1], -1.0f), 1.0f);
      lo.z = fminf(fmaxf(accY[2], -1.0f), 1.0f);
      lo.w = fminf(fmaxf(accY[3], -1.0f), 1.0f);
      hi.x = fminf(fmaxf(accY[4], -1.0f), 1.0f);
      hi.y = fminf(fmaxf(accY[5], -1.0f), 1.0f);
      hi.z = fminf(fmaxf(accY[6], -1.0f), 1.0f);
      hi.w = fminf(fmaxf(accY[7], -1.0f), 1.0f);
      *(float4*)(o)     = lo;
      *(float4*)(o + 4) = hi;
    }

    // commit new state: lane L<16 holds full 8-state of channel L contiguously
    if (lane < 16) {
#pragma unroll
      for (int k = 0; k < 8; ++k) Zlds[nIdx * 8 + k] = accZ[k];
    }
  }
}

extern "C" void kernel_launch(void* const* d_in, const int* in_sizes, int n_in,
                              void* d_out, int out_size, void* d_ws, size_t ws_size,
                              hipStream_t stream) {
  const float* x = (const float*)d_in[0];
  const float* A = (const float*)d_in[1];
  const float* B = (const float*)d_in[2];
  float* out = (float*)d_out;
  float* ws  = (float*)d_ws;

  iir_setup_kernel<<<1, 1, 0, stream>>>(A, B, ws);

  const int totalWaves = CGRPS * NSEG;              // 2048
  const int blocks = totalWaves / WPB;              // 256
  iir_wmma_kernel<<<blocks, WPB * 32, 0, stream>>>(x, ws, out);
}